// MemoryModule_41841571397801
// MI455X (gfx1250) — compile-verified
//
#include <hip/hip_runtime.h>
#include <math.h>

typedef __attribute__((ext_vector_type(2))) float v2f;
typedef __attribute__((ext_vector_type(8))) float v8f;

#define B_ROWS 32
#define M_ROWS 10000
#define D_DIM  512
#define TOPK   5
#define RCUBE  32768   /* 32*32*32 */

/* workspace layout (float index units) */
#define WS_INVNX 0
#define WS_INVNK 32
#define WS_SIM   10048                 /* 16-aligned; 32*10000 floats */
#define WS_IDX   (10048 + 320000)      /* 160 ints live here */

/* ---------------- row L2-norm reciprocal: one wave per row ---------------- */
__global__ void rownorm_kernel(const float* __restrict__ A,
                               float* __restrict__ inv,
                               int rows_per_block) {
  const int lane = threadIdx.x & 31;
  const int wave = threadIdx.x >> 5;
  const int row  = blockIdx.x * rows_per_block + wave;
  const float4* p = (const float4*)(A + (size_t)row * D_DIM);
  float s = 0.0f;
#pragma unroll
  for (int i = 0; i < 4; ++i) {               /* 512 floats = 128 float4, 4 per lane */
    float4 v = p[lane + 32 * i];
    s += v.x * v.x + v.y * v.y + v.z * v.z + v.w * v.w;
  }
#pragma unroll
  for (int off = 16; off > 0; off >>= 1) s += __shfl_xor(s, off, 32);
  if (lane == 0) inv[row] = 1.0f / fmaxf(sqrtf(s), 1e-12f);
}

/* ------- cosine-sim GEMM: one wave -> one 16x16 tile via f32 WMMA --------- */
/* sim[b][m] = dot(x[b],key[m]) * invnx[b] * invnk[m]                         */
__global__ __launch_bounds__(32)
void sim_wmma_kernel(const float* __restrict__ X,
                     const float* __restrict__ K,
                     const float* __restrict__ invnx,
                     const float* __restrict__ invnk,
                     float* __restrict__ sim) {
  const int lane = threadIdx.x;          /* 0..31 */
  const int n0   = blockIdx.x * 16;      /* memory-slot tile */
  const int b0   = blockIdx.y * 16;      /* batch tile       */
  const int half = lane >> 4;            /* 0: K=0,1  1: K=2,3  (ISA 16x4 f32 layout) */
  const int r    = lane & 15;            /* A row / B col within tile */

  const float* xrow = X + (size_t)(b0 + r) * D_DIM + half * 2;
  const float* krow = K + (size_t)(n0 + r) * D_DIM + half * 2;

  v8f acc = {};
#pragma unroll 4
  for (int d = 0; d < D_DIM; d += 4) {
    v2f a = *(const v2f*)(xrow + d);     /* A[m][k], A[m][k+1] */
    v2f b = *(const v2f*)(krow + d);     /* B[k][n], B[k+1][n] (= key[n][d+k..]) */
    acc = __builtin_amdgcn_wmma_f32_16x16x4_f32(
        /*neg_a=*/false, a, /*neg_b=*/false, b,
        /*c_mod=*/(short)0, acc, /*reuse_a=*/false, /*reuse_b=*/false);
  }

  /* C/D layout: value j in lane -> row M = j + 8*(lane>=16), col N = lane%16 */
  const float sk = invnk[n0 + r];
#pragma unroll
  for (int j = 0; j < 8; ++j) {
    const int m = b0 + j + 8 * half;
    sim[(size_t)m * M_ROWS + n0 + r] = acc[j] * invnx[m] * sk;
  }
}

/* ---------------- top-5 per batch row: one wave per row ------------------- */
__global__ void topk_kernel(const float* __restrict__ sim,
                            float* __restrict__ out_idx_f,
                            float* __restrict__ out_vals,
                            int*   __restrict__ idx_ws) {
  __shared__ float sv[32 * TOPK];
  __shared__ int   si[32 * TOPK];
  const int b    = blockIdx.x;
  const int lane = threadIdx.x;
  const float* row = sim + (size_t)b * M_ROWS;

  float bv[TOPK];
  int   bi[TOPK];
#pragma unroll
  for (int i = 0; i < TOPK; ++i) { bv[i] = -INFINITY; bi[i] = 0x7fffffff; }

  for (int m = lane; m < M_ROWS; m += 32) {
    const float v = row[m];
    if (v > bv[TOPK - 1]) {
      int p = TOPK - 1;
      while (p > 0 && v > bv[p - 1]) { bv[p] = bv[p - 1]; bi[p] = bi[p - 1]; --p; }
      bv[p] = v; bi[p] = m;            /* strict '>' keeps lowest index on ties */
    }
  }
#pragma unroll
  for (int i = 0; i < TOPK; ++i) { sv[lane * TOPK + i] = bv[i]; si[lane * TOPK + i] = bi[i]; }
  __syncthreads();

  if (lane == 0) {
    for (int k = 0; k < TOPK; ++k) {
      float best = -INFINITY; int besti = 0x7fffffff; int bestp = 0;
      for (int t = 0; t < 32 * TOPK; ++t) {
        const float v = sv[t];
        if (v > best || (v == best && si[t] < besti)) { best = v; besti = si[t]; bestp = t; }
      }
      sv[bestp] = -INFINITY;
      out_vals[b * TOPK + k]  = best;
      out_idx_f[b * TOPK + k] = (float)besti;
      idx_ws[b * TOPK + k]    = besti;
    }
  }
}

/* --------- gather voxel grids: (b,k) slot -> 128KB streaming copy --------- */
__global__ void gather_kernel(const float* __restrict__ vals,
                              const int*   __restrict__ idx,
                              float* __restrict__ out) {
  const int slot = blockIdx.y;                               /* 0..159 */
  const int t    = blockIdx.x * blockDim.x + threadIdx.x;    /* 0..8191 float4 */
  const float4* src = (const float4*)(vals + (size_t)idx[slot] * RCUBE);
  float4*       dst = (float4*)(out + (size_t)slot * RCUBE);
  dst[t] = src[t];
}

extern "C" void kernel_launch(void* const* d_in, const int* in_sizes, int n_in,
                              void* d_out, int out_size, void* d_ws, size_t ws_size,
                              hipStream_t stream) {
  const float* x    = (const float*)d_in[0];   /* (32, 512)        */
  const float* keys = (const float*)d_in[1];   /* (10000, 512)     */
  const float* vals = (const float*)d_in[2];   /* (10000, 32,32,32)*/
  /* d_in[3] = top_k scalar (==5, baked into TOPK) */
  (void)in_sizes; (void)n_in; (void)out_size; (void)ws_size;

  float* ws     = (float*)d_ws;
  float* invnx  = ws + WS_INVNX;
  float* invnk  = ws + WS_INVNK;
  float* sim    = ws + WS_SIM;
  int*   idx_ws = (int*)(ws + WS_IDX);

  float* out_ret   = (float*)d_out;                                  /* (32,5,32,32,32) */
  float* out_idx_f = out_ret + (size_t)B_ROWS * TOPK * RCUBE;        /* 160 floats      */
  float* out_vals  = out_idx_f + B_ROWS * TOPK;                      /* 160 floats      */

  /* 1) inverse norms */
  rownorm_kernel<<<dim3(B_ROWS), dim3(32), 0, stream>>>(x, invnx, 1);
  rownorm_kernel<<<dim3(M_ROWS / 8), dim3(256), 0, stream>>>(keys, invnk, 8);

  /* 2) cosine similarity GEMM via f32 WMMA: 625 N-tiles x 2 M-tiles */
  sim_wmma_kernel<<<dim3(M_ROWS / 16, B_ROWS / 16), dim3(32), 0, stream>>>(
      x, keys, invnx, invnk, sim);

  /* 3) top-5 per row */
  topk_kernel<<<dim3(B_ROWS), dim3(32), 0, stream>>>(sim, out_idx_f, out_vals, idx_ws);

  /* 4) gather voxel grids (bandwidth-dominant: ~42 MB r+w) */
  gather_kernel<<<dim3(RCUBE / 4 / 256, B_ROWS * TOPK), dim3(256), 0, stream>>>(
      vals, idx_ws, out_ret);
}